// Restricted_SelfAttention_89069031784946
// MI455X (gfx1250) — compile-verified
//
#include <hip/hip_runtime.h>
#include <math.h>

typedef float v2f __attribute__((ext_vector_type(2)));
typedef float v8f __attribute__((ext_vector_type(8)));

#define BATCH     64
#define SEQ       8192
#define NUM_HEADS 8
#define HIDDEN    256
#define QK_HEAD   16
#define V_HEAD    32

// ---------------------------------------------------------------------------
// Kernel 1: per-batch prep.
//   - block-reduce min/max of x[b, :]
//   - a[b,h] = scale * sum_d (x[b,0]*q_w + q_b) * k_w       (k_b cancels in softmax)
//   - m[b,h] = a>0 ? a*max(x) : a*min(x)    (exact softmax max, no online scan)
// ---------------------------------------------------------------------------
__global__ void prep_kernel(const float* __restrict__ x,
                            const float* __restrict__ qw,
                            const float* __restrict__ qb,
                            const float* __restrict__ kw,
                            float* __restrict__ a_ws,
                            float* __restrict__ m_ws) {
    __shared__ float smax[256];
    __shared__ float smin[256];
    const int b   = blockIdx.x;
    const int tid = threadIdx.x;
    const float* xb = x + (size_t)b * SEQ;

    float mx = -INFINITY, mn = INFINITY;
    for (int t = tid; t < SEQ; t += 256) {
        float v = xb[t];
        mx = fmaxf(mx, v);
        mn = fminf(mn, v);
    }
    smax[tid] = mx; smin[tid] = mn;
    __syncthreads();
    for (int s = 128; s > 0; s >>= 1) {
        if (tid < s) {
            smax[tid] = fmaxf(smax[tid], smax[tid + s]);
            smin[tid] = fminf(smin[tid], smin[tid + s]);
        }
        __syncthreads();
    }

    if (tid < NUM_HEADS) {
        const float x0 = xb[0];
        float alpha = 0.f;
        #pragma unroll
        for (int d = 0; d < QK_HEAD; ++d) {
            const int i = tid * QK_HEAD + d;
            alpha += fmaf(x0, qw[i], qb[i]) * kw[i];
        }
        const float a = alpha * 0.17677669529663688f;   // 1/sqrt(V_HEAD=32)
        const float m = (a > 0.f) ? a * smax[0] : a * smin[0];
        a_ws[b * NUM_HEADS + tid] = a;
        m_ws[b * NUM_HEADS + tid] = m;
    }
}

// ---------------------------------------------------------------------------
// Kernel 2: attention context via V_WMMA_F32_16X16X4_F32 (fp32 matrix core).
//   grid = (8 heads, 4 batch-tiles), one wave32 per block (EXEC all-1s).
//   For each head: ctx_tile(16b x 32d) = sum_t E[b,t] * G[t, h*32+d],
//   E[b,t] = exp(a*x - m) * x; Z accumulated in VALU, reduced via shfl_xor(16).
//
//   A (16x4 f32) lane layout: lanes 0-15 rows M, K={0,1}; lanes 16-31 K={2,3}.
//   B (4x16 f32) lane layout: VGPR0 rows {0|2}, VGPR1 rows {1|3}, N = lane&15.
// ---------------------------------------------------------------------------
__global__ void attn_wmma_kernel(const float* __restrict__ x,
                                 const float* __restrict__ G,
                                 const float* __restrict__ a_ws,
                                 const float* __restrict__ m_ws,
                                 float* __restrict__ ctx) {
    const int h    = blockIdx.x;      // head 0..7
    const int bt   = blockIdx.y;      // batch tile 0..3
    const int lane = threadIdx.x;     // 0..31 (wave32)
    const int M    = lane & 15;       // A row / B column within tile
    const int hi   = lane >> 4;       // K-pair select: 0 -> K{0,1}, 1 -> K{2,3}
    const int b    = bt * 16 + M;
    const int c0   = h * V_HEAD;      // column base in HIDDEN

    const float a = a_ws[b * NUM_HEADS + h];
    const float m = m_ws[b * NUM_HEADS + h];
    const float* xb = x + (size_t)b * SEQ;

    v8f acc0 = {};
    v8f acc1 = {};
    float zacc = 0.f;

    for (int t0 = 0; t0 < SEQ; t0 += 4) {
        const int k = t0 + hi * 2;
        // two consecutive x values for this lane's K pair (8B aligned)
        const v2f xv = *(const v2f*)(xb + k);
        const float e0 = __expf(fmaf(a, xv.x, -m));
        const float e1 = __expf(fmaf(a, xv.y, -m));
        zacc += e0 + e1;

        v2f A;
        A.x = e0 * xv.x;
        A.y = e1 * xv.y;

        // B tiles: rows k, k+1 ; columns c0 + {0,16} + (lane&15)
        const float* g0 = G + (size_t)k * HIDDEN + c0 + M;
        const float* g1 = g0 + HIDDEN;
        v2f B0; B0.x = g0[0];  B0.y = g1[0];
        v2f B1; B1.x = g0[16]; B1.y = g1[16];

        acc0 = __builtin_amdgcn_wmma_f32_16x16x4_f32(false, A, false, B0,
                                                     (short)0, acc0, false, false);
        acc1 = __builtin_amdgcn_wmma_f32_16x16x4_f32(false, A, false, B1,
                                                     (short)0, acc1, false, false);
    }

    // Z: lanes L and L^16 cover complementary K phases of the same batch row.
    const float z = zacc + __shfl_xor(zacc, 16, 32);
    // After this, lane L holds full Z for batch row (L & 15).

    // C layout: VGPR v -> row (v + 8*hi), N = lane&15.
    #pragma unroll
    for (int v = 0; v < 8; ++v) {
        const int   r   = v + 8 * hi;              // row within 16-row tile
        const float zr  = __shfl(z, r, 32);        // Z for that batch row
        const int   row = bt * 16 + r;             // global batch index
        const int   col = c0 + M;
        ctx[row * HIDDEN + col]      = acc0[v] / zr;
        ctx[row * HIDDEN + col + 16] = acc1[v] / zr;
    }
}

// ---------------------------------------------------------------------------
// Kernel 3: descending sort of each 256-vector (rank-by-count, stable).
// ---------------------------------------------------------------------------
__global__ void sort_kernel(const float* __restrict__ ctx,
                            float* __restrict__ out) {
    __shared__ float s[HIDDEN];
    const int b   = blockIdx.x;
    const int tid = threadIdx.x;
    const float v = ctx[b * HIDDEN + tid];
    s[tid] = v;
    __syncthreads();
    int rank = 0;
    for (int j = 0; j < HIDDEN; ++j) {
        const float u = s[j];
        rank += (u > v) || (u == v && j < tid);
    }
    out[b * HIDDEN + rank] = v;
}

// ---------------------------------------------------------------------------
extern "C" void kernel_launch(void* const* d_in, const int* in_sizes, int n_in,
                              void* d_out, int out_size, void* d_ws, size_t ws_size,
                              hipStream_t stream) {
    const float* x  = (const float*)d_in[0];   // (64, 8192, 1)
    const float* qw = (const float*)d_in[1];   // (128, 1)
    const float* qb = (const float*)d_in[2];   // (128,)
    const float* kw = (const float*)d_in[3];   // (128, 1)
    // d_in[4] = k_b : unused (cancels in softmax)
    const float* G  = (const float*)d_in[5];   // (8192, 256)
    float* out = (float*)d_out;                // (64, 1, 256)

    float* wsf  = (float*)d_ws;
    float* a_ws = wsf;                 // 64*8
    float* m_ws = wsf + 512;           // 64*8
    float* ctx  = wsf + 1024;          // 64*256

    prep_kernel<<<BATCH, 256, 0, stream>>>(x, qw, qb, kw, a_ws, m_ws);

    dim3 grid(NUM_HEADS, BATCH / 16);
    attn_wmma_kernel<<<grid, 32, 0, stream>>>(x, G, a_ws, m_ws, ctx);

    sort_kernel<<<BATCH, HIDDEN, 0, stream>>>(ctx, out);
}